// Encoder_23639499997815
// MI455X (gfx1250) — compile-verified
//
#include <hip/hip_runtime.h>

#define IN_DIM  256
#define HID_DIM 128
#define OUT_DIM 64

typedef __attribute__((ext_vector_type(2))) float v2f;
typedef __attribute__((ext_vector_type(8))) float v8f;

// ---------------------------------------------------------------- degree prep
__global__ void init_deg_kernel(float* __restrict__ deg, int n) {
  int i = blockIdx.x * blockDim.x + threadIdx.x;
  if (i < n) deg[i] = 1.0f;                       // self-loop contribution
}

__global__ void accum_deg_kernel(const int* __restrict__ dst,
                                 float* __restrict__ deg, int e) {
  int i = blockIdx.x * blockDim.x + threadIdx.x;
  if (i < e) atomicAdd(&deg[dst[i]], 1.0f);
}

__global__ void dinv_kernel(float* __restrict__ deg, int n) {
  int i = blockIdx.x * blockDim.x + threadIdx.x;
  if (i < n) deg[i] = rsqrtf(deg[i]);             // deg >= 1 always (self loops)
}

// ------------------------------------------------------- fp32 WMMA dense GEMM
// C[M,N] = A[M,K] @ B[K,N], N = NT*16.  One wave32 per 16-row strip, NT col
// tiles accumulated in registers via V_WMMA_F32_16X16X4_F32.
template <int K, int NT>
__global__ void gemm_wmma_kernel(const float* __restrict__ A,
                                 const float* __restrict__ B,
                                 float* __restrict__ C, int M) {
  const int N = NT * 16;
  const int wave = (int)((blockIdx.x * (unsigned)blockDim.x + threadIdx.x) >> 5);
  const int lane = threadIdx.x & 31;
  const int row0 = wave * 16;
  if (row0 >= M) return;                          // wave-uniform exit

  const int mn    = lane & 15;                    // M index for A, N index for B/C
  const int khalf = lane >> 4;                    // selects K pair {0,1} vs {2,3}

  int arow = row0 + mn;
  if (arow >= M) arow = M - 1;                    // clamp (keeps EXEC all-ones)
  const float* __restrict__ Ar = A + (size_t)arow * K;

  v8f acc[NT];
#pragma unroll
  for (int t = 0; t < NT; ++t) acc[t] = v8f{};

  for (int k = 0; k < K; k += 4) {
    const int ka = k + 2 * khalf;
    v2f a;                                        // A 16x4 tile: vgpr = k%2
    a.x = Ar[ka];
    a.y = Ar[ka + 1];
#pragma unroll
    for (int t = 0; t < NT; ++t) {
      v2f b;                                      // B 4x16 tile: lane = n + 16*(k/2)
      b.x = B[(size_t)ka       * N + t * 16 + mn];
      b.y = B[(size_t)(ka + 1) * N + t * 16 + mn];
      acc[t] = __builtin_amdgcn_wmma_f32_16x16x4_f32(
          /*neg_a=*/false, a, /*neg_b=*/false, b,
          /*c_mod=*/(short)0, acc[t], /*reuse_a=*/false, /*reuse_b=*/false);
    }
  }

  // C/D layout: VGPR i -> M = i + 8*(lane>=16), N = lane&15
#pragma unroll
  for (int t = 0; t < NT; ++t) {
#pragma unroll
    for (int i = 0; i < 8; ++i) {
      const int row = row0 + 8 * khalf + i;
      if (row < M) C[(size_t)row * N + t * 16 + mn] = acc[t][i];
    }
  }
}

// ------------------------------------------------------------- graph scatter
__global__ void init_bias_kernel(float* __restrict__ out,
                                 const float* __restrict__ bias,
                                 int fmask, long long total) {
  long long idx = blockIdx.x * (long long)blockDim.x + threadIdx.x;
  if (idx < total) out[idx] = bias[(int)(idx & fmask)];
}

// agg[dst] += h[src] * dinv[src]*dinv[dst]; edges e >= n_edges are self loops.
template <int F>
__global__ void scatter_kernel(const float* __restrict__ h,
                               const int* __restrict__ src,
                               const int* __restrict__ dst,
                               const float* __restrict__ dinv,
                               float* __restrict__ agg,
                               int n_edges, int n_nodes) {
  long long idx = blockIdx.x * (long long)blockDim.x + threadIdx.x;
  long long total = (long long)(n_edges + n_nodes) * F;
  if (idx >= total) return;
  const int e = (int)(idx / F);
  const int f = (int)(idx & (F - 1));
  int s, d;
  if (e < n_edges) { s = src[e]; d = dst[e]; }
  else             { s = d = e - n_edges; }       // self loop
  const float norm = dinv[s] * dinv[d];
  atomicAdd(&agg[(size_t)d * F + f], h[(size_t)s * F + f] * norm);
}

__global__ void relu_kernel(float* __restrict__ v, long long total) {
  long long idx = blockIdx.x * (long long)blockDim.x + threadIdx.x;
  if (idx < total) {
    float x = v[idx];
    v[idx] = x > 0.0f ? x : 0.0f;
  }
}

// ------------------------------------------------------------------ launcher
extern "C" void kernel_launch(void* const* d_in, const int* in_sizes, int n_in,
                              void* d_out, int out_size, void* d_ws, size_t ws_size,
                              hipStream_t stream) {
  const float* x  = (const float*)d_in[0];
  const int*   ei = (const int*)d_in[1];
  const float* W1 = (const float*)d_in[2];
  const float* b1 = (const float*)d_in[3];
  const float* W2 = (const float*)d_in[4];
  const float* b2 = (const float*)d_in[5];
  float* out = (float*)d_out;

  const int N = in_sizes[0] / IN_DIM;             // 50000
  const int E = in_sizes[1] / 2;                  // 800000
  const int* src = ei;                            // edge_index[0]
  const int* dst = ei + E;                        // edge_index[1]

  // workspace layout (floats): [dinv: N][h: N*HID (reused as h2: N*OUT)][agg1: N*HID]
  float* dinv = (float*)d_ws;
  float* h    = dinv + N;
  float* agg1 = h + (size_t)N * HID_DIM;

  const int B256 = 256;
  // ---- symmetric normalization
  init_deg_kernel<<<(N + B256 - 1) / B256, B256, 0, stream>>>(dinv, N);
  accum_deg_kernel<<<(E + B256 - 1) / B256, B256, 0, stream>>>(dst, dinv, E);
  dinv_kernel<<<(N + B256 - 1) / B256, B256, 0, stream>>>(dinv, N);

  const int waves  = (N + 15) / 16;               // one wave per 16-row strip
  const int gblks  = (waves + 3) / 4;             // 128 threads = 4 waves/block

  // ---- layer 1: h = x @ W1 ; agg1 = scatter(h)+b1 ; relu in place
  gemm_wmma_kernel<IN_DIM, HID_DIM / 16><<<gblks, 128, 0, stream>>>(x, W1, h, N);
  const long long tot1 = (long long)N * HID_DIM;
  init_bias_kernel<<<(int)((tot1 + B256 - 1) / B256), B256, 0, stream>>>(agg1, b1, HID_DIM - 1, tot1);
  const long long sc1 = (long long)(E + N) * HID_DIM;
  scatter_kernel<HID_DIM><<<(int)((sc1 + B256 - 1) / B256), B256, 0, stream>>>(
      h, src, dst, dinv, agg1, E, N);
  relu_kernel<<<(int)((tot1 + B256 - 1) / B256), B256, 0, stream>>>(agg1, tot1);

  // ---- layer 2: h2 = relu @ W2 (reuse h buffer) ; out = scatter(h2)+b2
  gemm_wmma_kernel<HID_DIM, OUT_DIM / 16><<<gblks, 128, 0, stream>>>(agg1, W2, h, N);
  const long long tot2 = (long long)N * OUT_DIM;
  init_bias_kernel<<<(int)((tot2 + B256 - 1) / B256), B256, 0, stream>>>(out, b2, OUT_DIM - 1, tot2);
  const long long sc2 = (long long)(E + N) * OUT_DIM;
  scatter_kernel<OUT_DIM><<<(int)((sc2 + B256 - 1) / B256), B256, 0, stream>>>(
      h, src, dst, dinv, out, E, N);
}